// SwappedPredictionLoss_1726576856935
// MI455X (gfx1250) — compile-verified
//
#include <hip/hip_runtime.h>

typedef float v2f __attribute__((ext_vector_type(2)));
typedef float v8f __attribute__((ext_vector_type(8)));

#define BATCH   4096
#define KPROT   3000
#define KPAD    3072            // padded K (u[k>=3000] = 0)
#define INV_EPS 20.0f           // 1/0.05
#define INV_T   10.0f           // 1/0.1
#define KF      3000.0f
#define BF      4096.0f

// workspace layout (floats)
#define OFF_U        0          // 3072 used
#define OFF_V        4096       // 4096 used
#define OFF_CONTRIB  8192       // 4096 used

// ---------------------------------------------------------------- utilities
__global__ void swav_zero_out(float* out) { out[0] = 0.0f; }

__global__ void swav_init_v(float* ws) {
    int b = blockIdx.x * blockDim.x + threadIdx.x;
    if (b < BATCH) ws[OFF_V + b] = 1.0f;
}

// ------------------------------------------------------ u[k] = 1/(K * sum_b E[b,k] * v[b])
// 96 blocks x 256 threads. Each wave owns one b-slice of 512 rows, 32 consecutive k
// per wave => coalesced 128B row segments. Fixed-order LDS combine across 8 waves.
__global__ void swav_colsum_u(const float* __restrict__ scores,
                              const long long* __restrict__ code_ids,
                              int pair, float* __restrict__ ws) {
    __shared__ float red[8][33];
    const int sid = (int)code_ids[pair];
    const float* S = scores + (size_t)sid * BATCH * KPROT;
    const float* v = ws + OFF_V;
    const int lane = threadIdx.x & 31;
    const int w    = threadIdx.x >> 5;
    const int k    = blockIdx.x * 32 + lane;
    float acc = 0.0f;
    if (k < KPROT) {
        const int bstart = w * (BATCH / 8);
        for (int b = bstart; b < bstart + (BATCH / 8); ++b)
            acc += __expf(S[(size_t)b * KPROT + k] * INV_EPS) * v[b];
    }
    red[w][lane] = acc;
    __syncthreads();
    if (threadIdx.x < 32) {
        int kk = blockIdx.x * 32 + threadIdx.x;
        float s = 0.0f;
        #pragma unroll
        for (int j = 0; j < 8; ++j) s += red[j][threadIdx.x];   // deterministic order
        float u = (kk < KPROT) ? (1.0f / (KF * s)) : 0.0f;      // zero-pad tail
        ws[OFF_U + kk] = u;
    }
}

// ------------------------------------------------------ v[b] = 1/(B * sum_k E[b,k] * u[k])
// WMMA engine: 256 blocks (16 rows each) x 128 threads (4 waves).
// Tile E through LDS (pitch 132 -> conflict-free 16-row column reads), then
// D(16x16) += A(16x4: exp values) x B(4x16: u[k] broadcast) accumulates the
// u-weighted row sums exactly in f32; row sums extracted from lanes 0/16.
__global__ void swav_rowsum_v_wmma(const float* __restrict__ scores,
                                   const long long* __restrict__ code_ids,
                                   int pair, float* __restrict__ ws) {
    __shared__ float tileA[16][132];
    __shared__ float part[4][16];
    const int sid = (int)code_ids[pair];
    const float* S = scores + (size_t)sid * BATCH * KPROT;
    const float* u = ws + OFF_U;
    const int tid  = threadIdx.x;
    const int lane = tid & 31;
    const int w    = tid >> 5;
    const int b0   = blockIdx.x * 16;
    const int r    = lane & 15;
    const int hi   = (lane >> 4) << 1;       // 0 for lanes 0-15 (K0/K1), 2 for 16-31 (K2/K3)
    v8f c = {};
    for (int chunk = 0; chunk < KPAD / 128; ++chunk) {
        const int k0 = chunk * 128;
        // cooperative, coalesced load + exp into LDS (zero-fill past K)
        for (int idx = tid; idx < 16 * 128; idx += 128) {
            int row = idx >> 7, col = idx & 127;
            int k = k0 + col;
            float val = 0.0f;
            if (k < KPROT) val = __expf(S[(size_t)(b0 + row) * KPROT + k] * INV_EPS);
            tileA[row][col] = val;
        }
        __syncthreads();
        const int cbase = w * 32;            // each wave sweeps 32 columns (8 WMMAs)
        #pragma unroll
        for (int kk = 0; kk < 8; ++kk) {
            const int kc = cbase + kk * 4;
            v2f a, b;
            a.x = tileA[r][kc + hi];
            a.y = tileA[r][kc + hi + 1];
            b.x = u[k0 + kc + hi];           // B[K][n] = u broadcast over n
            b.y = u[k0 + kc + hi + 1];
            c = __builtin_amdgcn_wmma_f32_16x16x4_f32(false, a, false, b,
                                                      (short)0, c, false, false);
        }
        __syncthreads();
    }
    // D layout: lanes 0-15 hold M = vgpr, lanes 16-31 hold M = 8 + vgpr (any N column)
    if (lane == 0) {
        #pragma unroll
        for (int j = 0; j < 8; ++j) part[w][j] = c[j];
    }
    if (lane == 16) {
        #pragma unroll
        for (int j = 0; j < 8; ++j) part[w][8 + j] = c[j];
    }
    __syncthreads();
    if (tid < 16) {
        float sc = part[0][tid] + part[1][tid] + part[2][tid] + part[3][tid];
        ws[OFF_V + b0 + tid] = 1.0f / (BF * sc);
    }
}

// ------------------------------------------------------ per-row loss contribution
// contrib[b] = lse_b/B - v[b] * sum_k E_s[b,k]*u[k]*(s_t[b,k]/T)
__global__ void swav_loss_contrib(const float* __restrict__ scores,
                                  const long long* __restrict__ code_ids,
                                  const long long* __restrict__ score_ids,
                                  int pair, float* __restrict__ ws) {
    __shared__ float sm[256], sd[256], sa[256];
    const int sid = (int)code_ids[pair];
    const int tvw = (int)score_ids[pair];
    const int b   = blockIdx.x;
    const float* Ss = scores + (size_t)sid * BATCH * KPROT + (size_t)b * KPROT;
    const float* St = scores + (size_t)tvw * BATCH * KPROT + (size_t)b * KPROT;
    const float* u  = ws + OFF_U;
    float m = -1.0e30f, d = 0.0f, a = 0.0f;
    for (int k = threadIdx.x; k < KPROT; k += 256) {
        float st = St[k] * INV_T;
        if (st > m) { d = d * __expf(m - st) + 1.0f; m = st; }
        else        { d += __expf(st - m); }
        a += __expf(Ss[k] * INV_EPS) * u[k] * st;
    }
    sm[threadIdx.x] = m; sd[threadIdx.x] = d; sa[threadIdx.x] = a;
    __syncthreads();
    for (int off = 128; off > 0; off >>= 1) {
        if (threadIdx.x < off) {
            float m1 = sm[threadIdx.x],       d1 = sd[threadIdx.x];
            float m2 = sm[threadIdx.x + off], d2 = sd[threadIdx.x + off];
            float M  = fmaxf(m1, m2);
            sd[threadIdx.x] = d1 * __expf(m1 - M) + d2 * __expf(m2 - M);
            sm[threadIdx.x] = M;
            sa[threadIdx.x] += sa[threadIdx.x + off];
        }
        __syncthreads();
    }
    if (threadIdx.x == 0) {
        float lse = sm[0] + __logf(sd[0]);
        ws[OFF_CONTRIB + b] = lse * (1.0f / BF) - ws[OFF_V + b] * sa[0];
    }
}

// ------------------------------------------------------ deterministic scalar reduce
__global__ void swav_reduce_add(const float* __restrict__ ws, float* out) {
    __shared__ float s[256];
    float acc = 0.0f;
    for (int i = threadIdx.x; i < BATCH; i += 256) acc += ws[OFF_CONTRIB + i];
    s[threadIdx.x] = acc;
    __syncthreads();
    for (int off = 128; off > 0; off >>= 1) {
        if (threadIdx.x < off) s[threadIdx.x] += s[threadIdx.x + off];
        __syncthreads();
    }
    if (threadIdx.x == 0) out[0] += s[0];
}

// ---------------------------------------------------------------- launcher
extern "C" void kernel_launch(void* const* d_in, const int* in_sizes, int n_in,
                              void* d_out, int out_size, void* d_ws, size_t ws_size,
                              hipStream_t stream) {
    const float*     scores    = (const float*)d_in[0];
    const long long* code_ids  = (const long long*)d_in[1];
    const long long* score_ids = (const long long*)d_in[2];
    float* out = (float*)d_out;
    float* ws  = (float*)d_ws;

    swav_zero_out<<<1, 1, 0, stream>>>(out);
    for (int pair = 0; pair < 2; ++pair) {
        swav_init_v<<<BATCH / 256, 256, 0, stream>>>(ws);
        for (int it = 0; it < 3; ++it) {   // n_iters = 3 (fixed, as in setup_inputs)
            swav_colsum_u<<<KPAD / 32, 256, 0, stream>>>(scores, code_ids, pair, ws);
            swav_rowsum_v_wmma<<<BATCH / 16, 128, 0, stream>>>(scores, code_ids, pair, ws);
        }
        swav_loss_contrib<<<BATCH, 256, 0, stream>>>(scores, code_ids, score_ids, pair, ws);
        swav_reduce_add<<<1, 256, 0, stream>>>(ws, out);
    }
}